// RGBBlock_1365799600525
// MI455X (gfx1250) — compile-verified
//
#include <hip/hip_runtime.h>
#include <hip/hip_bf16.h>

typedef __attribute__((ext_vector_type(2))) float v2f;
typedef __attribute__((ext_vector_type(8))) float v8f;

#define BATCH 16
#define C_IN  512
#define OUTC  3
#define HDIM  128
#define HW    (HDIM * HDIM)      // 16384
#define ODIM  256

// ---------------------------------------------------------------------------
// Stage 1: modulated 1x1 conv as WMMA f32 16x16x4 GEMM.
//   out0[b][o][hw] = sum_k ( W[o][k] * (y[b][k]+1) ) * x[b][k][hw]
// Wave: 16-pixel tile, full K=512 reduction, D = A(x tile 16x4) * B(Mw 4x16).
// A layout (32-bit, 16x4): lane l -> pixel = l&15; VGPR0 holds K = 2*(l>>4),
// VGPR1 holds K = 2*(l>>4)+1 (ISA 7.12.2). B (4x16) mirrored: lane -> col,
// VGPR0/1 -> same K pair. D: VGPR r, lane l -> pixel r+8*(l>>4), col l&15.
// ---------------------------------------------------------------------------
__global__ void __launch_bounds__(256)
mod_conv_wmma(const float* __restrict__ x,
              const float* __restrict__ y,
              const float* __restrict__ W,
              float* __restrict__ out0) {
    __shared__ float lmw[C_IN * 16];          // [k][o16], padded cols 3..15 = 0

    const int b   = blockIdx.y;
    const int tid = threadIdx.x;

    // Build modulated, padded weight matrix in LDS (cheap: 2*3*512 cached loads).
    for (int idx = tid; idx < C_IN * 16; idx += 256) {
        const int k = idx >> 4;
        const int o = idx & 15;
        float v = 0.0f;
        if (o < OUTC) v = W[o * C_IN + k] * (y[b * C_IN + k] + 1.0f);
        lmw[idx] = v;
    }
    __syncthreads();

    const int wave = tid >> 5;
    const int lane = tid & 31;
    const int n    = lane & 15;               // pixel-in-tile (A) / out col (B,D)
    const int half = lane >> 4;
    const int tile = blockIdx.x * 8 + wave;   // 16-pixel tile in HW
    const int px   = tile * 16;

    const float* xb = x + ((size_t)b * C_IN) * HW + px;

    v8f acc = {};
    #pragma unroll 4
    for (int k0 = 0; k0 < C_IN; k0 += 4) {
        const int ka = k0 + 2 * half;
        v2f a, bm;
        a.x  = xb[(size_t)ka * HW + n];
        a.y  = xb[(size_t)(ka + 1) * HW + n];
        bm.x = lmw[ka * 16 + n];
        bm.y = lmw[(ka + 1) * 16 + n];
        // look ahead 16 channels (speculative, OOB-safe)
        __builtin_prefetch(xb + (size_t)(ka + 16) * HW + n, 0, 1);
        acc = __builtin_amdgcn_wmma_f32_16x16x4_f32(
                  false, a, false, bm, (short)0, acc, false, false);
    }

    if (n < OUTC) {
        float* op = out0 + ((size_t)b * OUTC + n) * HW + px + half * 8;
        #pragma unroll
        for (int r = 0; r < 8; ++r) op[r] = acc[r];
    }
}

// ---------------------------------------------------------------------------
// Stage 2: fused 2x bilinear upsample (half-pixel, edge clamp) + 3x3 [1,2,1]
// blur with reflect padding. One thread per output pixel; src is L2-resident.
// ---------------------------------------------------------------------------
__device__ __forceinline__ float bilin_up(const float* __restrict__ p, int Yu, int Xu) {
    float fy = Yu * 0.5f - 0.25f;
    float fx = Xu * 0.5f - 0.25f;
    int y0 = (int)floorf(fy);
    int x0 = (int)floorf(fx);
    float ty = fy - (float)y0;
    float tx = fx - (float)x0;
    int y1 = min(max(y0 + 1, 0), HDIM - 1);
    int x1 = min(max(x0 + 1, 0), HDIM - 1);
    y0 = min(max(y0, 0), HDIM - 1);
    x0 = min(max(x0, 0), HDIM - 1);
    float v0 = (1.0f - tx) * p[y0 * HDIM + x0] + tx * p[y0 * HDIM + x1];
    float v1 = (1.0f - tx) * p[y1 * HDIM + x0] + tx * p[y1 * HDIM + x1];
    return (1.0f - ty) * v0 + ty * v1;
}

__global__ void __launch_bounds__(256)
upsample_blur(const float* __restrict__ src, float* __restrict__ dst) {
    const size_t idx = (size_t)blockIdx.x * 256 + threadIdx.x;
    const int X  = (int)(idx & (ODIM - 1));
    const int Y  = (int)((idx >> 8) & (ODIM - 1));
    const int bc = (int)(idx >> 16);            // 0..47
    const float* p = src + (size_t)bc * HW;

    const float wgt[3] = {1.0f, 2.0f, 1.0f};
    float acc = 0.0f;
    #pragma unroll
    for (int dy = -1; dy <= 1; ++dy) {
        int Yu = Y + dy;
        if (Yu < 0) Yu = -Yu; else if (Yu > ODIM - 1) Yu = 2 * (ODIM - 1) - Yu;
        #pragma unroll
        for (int dx = -1; dx <= 1; ++dx) {
            int Xu = X + dx;
            if (Xu < 0) Xu = -Xu; else if (Xu > ODIM - 1) Xu = 2 * (ODIM - 1) - Xu;
            acc += wgt[dy + 1] * wgt[dx + 1] * bilin_up(p, Yu, Xu);
        }
    }
    dst[idx] = acc * (1.0f / 16.0f);
}

// ---------------------------------------------------------------------------
extern "C" void kernel_launch(void* const* d_in, const int* in_sizes, int n_in,
                              void* d_out, int out_size, void* d_ws, size_t ws_size,
                              hipStream_t stream) {
    const float* x = (const float*)d_in[0];   // (16, 512, 128, 128)
    const float* y = (const float*)d_in[1];   // (16, 512)
    const float* W = (const float*)d_in[2];   // (3, 512)
    float* out  = (float*)d_out;              // (16, 3, 256, 256)
    float* out0 = (float*)d_ws;               // (16, 3, 128, 128) scratch

    (void)in_sizes; (void)n_in; (void)out_size; (void)ws_size;

    // Stage 1: 8 waves/block * 16 pixels/wave = 128 pixels/block.
    dim3 grid1(HW / (16 * 8), BATCH);
    mod_conv_wmma<<<grid1, 256, 0, stream>>>(x, y, W, out0);

    // Stage 2: one thread per output pixel.
    const size_t total = (size_t)BATCH * OUTC * ODIM * ODIM;  // 3,145,728
    upsample_blur<<<(unsigned)(total / 256), 256, 0, stream>>>(out0, out);
}